// Decoder_20126216749091
// MI455X (gfx1250) — compile-verified
//
#include <hip/hip_runtime.h>

typedef float v2f __attribute__((ext_vector_type(2)));
typedef float v4f __attribute__((ext_vector_type(4)));
typedef float v8f __attribute__((ext_vector_type(8)));

#define DIMD 128
#define BATCH 512
#define SRCL 256
#define BT 8            // batch rows per workgroup (rows 8..15 of WMMA tile zero-padded)
#define NTHREADS 256
#define NWAVES 8
#define XS 132          // padded LDS stride for [16][DIM] arrays (avoids bank conflicts)
#define GS 516          // padded LDS stride for gate rows (4*DIM)
#define PS 260          // padded LDS stride for length-dim rows
#define C_EXPL 10.0f
#define LOG2E 1.44269504088896340736f
#define LN2 0.69314718055994530942f

// ---------------- CDNA5 primitives ----------------

__device__ __forceinline__ v8f wmma_f32_k4(v2f a, v2f b, v8f c) {
  // D(16x16,f32) = A(16x4,f32) x B(4x16,f32) + C
  return __builtin_amdgcn_wmma_f32_16x16x4_f32(false, a, false, b, (short)0, c,
                                               false, false);
}

__device__ __forceinline__ float tanh_fast(float x) {
  float y;
  // native CDNA5 transcendental; v_nop covers the 1-op TRANS result hazard
  asm volatile("v_tanh_f32 %0, %1\n\tv_nop" : "=v"(y) : "v"(x));
  return y;
}
__device__ __forceinline__ float exp2_fast(float x) { return __builtin_amdgcn_exp2f(x); }
__device__ __forceinline__ float log2_fast(float x) { return __builtin_amdgcn_logf(x); }
__device__ __forceinline__ float rcp_fast(float x)  { return __builtin_amdgcn_rcpf(x); }
__device__ __forceinline__ float sigmoid_fast(float x) {
  return rcp_fast(1.0f + exp2_fast(-LOG2E * x));
}

// ---------------- kernel 1: e = Wr @ ctx + br, layout e[b][l][d] ----------------
// One wave per 16x16 output tile: M = d_out, N = flattened (b*SRCL + l), K = d_in.
__global__ void __launch_bounds__(NTHREADS)
precompute_e_kernel(const float* __restrict__ Wr_g, const float* __restrict__ br_g,
                    const float* __restrict__ Wr_p, const float* __restrict__ br_p,
                    const float* __restrict__ context,
                    float* __restrict__ e_g, float* __restrict__ e_p) {
  const int lane = threadIdx.x & 31;
  const int gwave = blockIdx.x * (NTHREADS / 32) + (threadIdx.x >> 5);
  const int ctiles = (BATCH * SRCL) / 16;            // 8192 column tiles
  const int total = 2 * (DIMD / 16) * ctiles;        // 131072 waves
  if (gwave >= total) return;

  const int mat = gwave / ((DIMD / 16) * ctiles);
  int rem = gwave % ((DIMD / 16) * ctiles);
  const int mtile = rem / ctiles;
  const int ctile = rem % ctiles;

  const float* W  = mat ? Wr_p : Wr_g;
  const float* br = mat ? br_p : br_g;
  float*       e  = mat ? e_p  : e_g;

  const int mbase = mtile * 16;
  const int c0 = ctile * 16 + (lane & 15);           // flat column = b*SRCL + l
  const int bcol = c0 >> 8;                          // / SRCL
  const int lcol = c0 & (SRCL - 1);
  const int khalf = (lane >> 4) * 2;                 // lanes 16..31 hold K+2,K+3
  const int arow = mbase + (lane & 15);

  const float* wrow   = W + (size_t)arow * DIMD;
  const float* ctxcol = context + ((size_t)lcol * BATCH + bcol) * DIMD;

  v8f acc = {};
  for (int k = 0; k < DIMD; k += 4) {
    v2f a = *(const v2f*)(wrow + k + khalf);         // A[m][k..]
    v2f b = *(const v2f*)(ctxcol + k + khalf);       // B[k..][n] = ctx[l][b][k..]
    acc = wmma_f32_k4(a, b, acc);
  }
  // D: lane holds column c0; VGPR i -> M row (lane<16 ? i : 8+i)
  const int dbase = mbase + ((lane >> 4) << 3);
  float* dst = e + (size_t)c0 * DIMD + dbase;
#pragma unroll
  for (int i = 0; i < 8; ++i) dst[i] = acc[i] + br[dbase + i];
}

// ---------------- kernel 2: sequential pointer decoder ----------------
__global__ void __launch_bounds__(NTHREADS)
decoder_kernel(const float* __restrict__ decoder_input,
               const float* __restrict__ embedded_inputs,
               const float* __restrict__ h0, const float* __restrict__ c0,
               const float* __restrict__ W_ih, const float* __restrict__ W_hh,
               const float* __restrict__ b_ih, const float* __restrict__ b_hh,
               const float* __restrict__ Wq_g, const float* __restrict__ bq_g,
               const float* __restrict__ v_g,
               const float* __restrict__ Wq_p, const float* __restrict__ bq_p,
               const float* __restrict__ v_p,
               const float* __restrict__ e_g, const float* __restrict__ e_p,
               float* __restrict__ out_logp, float* __restrict__ out_sels,
               float* __restrict__ out_h, float* __restrict__ out_c) {
  __shared__ float sX[16 * XS], sH[16 * XS], sC[16 * XS], sQ[16 * XS], sG[16 * XS];
  __shared__ float sGate[BT * GS];
  __shared__ float sP[BT * PS];
  __shared__ float sVg[DIMD], sVp[DIMD];
  __shared__ unsigned sMask[BT][SRCL / 32];
  __shared__ int sIdx[BT];

  const int t = threadIdx.x;
  const int lane = t & 31;
  const int wave = t >> 5;
  const int bbase = blockIdx.x * BT;

  // ---- init: zero padding rows (so WMMA A-operands rows 8..15 contribute 0) ----
  for (int i = t; i < 16 * XS; i += NTHREADS) {
    sX[i] = 0.f; sH[i] = 0.f; sC[i] = 0.f; sQ[i] = 0.f; sG[i] = 0.f;
  }
  if (t < BT * (SRCL / 32)) sMask[t >> 3][t & 7] = 0u;
  if (t < BT) sIdx[t] = 0;
  if (t < DIMD) { sVg[t] = v_g[t]; sVp[t] = v_p[t]; }
  __syncthreads();
  for (int i = t; i < BT * DIMD; i += NTHREADS) {
    int m = i >> 7, d = i & (DIMD - 1);
    sX[m * XS + d] = decoder_input[(size_t)(bbase + m) * DIMD + d];
    sH[m * XS + d] = h0[(size_t)(bbase + m) * DIMD + d];
    sC[m * XS + d] = c0[(size_t)(bbase + m) * DIMD + d];
  }
  __syncthreads();

  const int arow = lane & 15;
  const int khalf = (lane >> 4) * 2;

  for (int step = 0; step < SRCL; ++step) {
    // ---- phase 0: fold previous selection into mask ----
    if (step > 0) {
      if (t < BT) {
        int pi = sIdx[t];
        sMask[t][pi >> 5] |= (1u << (pi & 31));
      }
    }
    __syncthreads();

    // ---- phase 1 (WMMA): gates = x@W_ih^T + h@W_hh^T + (b_ih+b_hh) ----
    {
#pragma unroll
      for (int nt = 0; nt < 4; ++nt) {
        const int col = (wave * 4 + nt) * 16 + arow;
        const float* w1 = W_ih + (size_t)col * DIMD;
        const float* w2 = W_hh + (size_t)col * DIMD;
        v8f acc = {};
        for (int k = 0; k < DIMD; k += 4) {
          v2f a = *(const v2f*)(&sX[arow * XS + k + khalf]);
          v2f b = *(const v2f*)(w1 + k + khalf);
          acc = wmma_f32_k4(a, b, acc);
        }
        for (int k = 0; k < DIMD; k += 4) {
          v2f a = *(const v2f*)(&sH[arow * XS + k + khalf]);
          v2f b = *(const v2f*)(w2 + k + khalf);
          acc = wmma_f32_k4(a, b, acc);
        }
        if (lane < 16) {                       // lanes<16 hold batch rows 0..7
          float bias = b_ih[col] + b_hh[col];
#pragma unroll
          for (int i = 0; i < 8; ++i) sGate[i * GS + col] = acc[i] + bias;
        }
      }
    }
    __syncthreads();

    // ---- phase 2: LSTM elementwise ----
    for (int i = t; i < BT * DIMD; i += NTHREADS) {
      int m = i >> 7, d = i & (DIMD - 1);
      float gi = sGate[m * GS + d];
      float gf = sGate[m * GS + DIMD + d];
      float gg = sGate[m * GS + 2 * DIMD + d];
      float go = sGate[m * GS + 3 * DIMD + d];
      float cn = sigmoid_fast(gf) * sC[m * XS + d] + sigmoid_fast(gi) * tanh_fast(gg);
      sC[m * XS + d] = cn;
      sH[m * XS + d] = sigmoid_fast(go) * tanh_fast(cn);
    }
    __syncthreads();

    // ---- phase 3 (WMMA): q = h @ Wq_g^T + bq_g ----
    {
      const int col = wave * 16 + arow;
      const float* w1 = Wq_g + (size_t)col * DIMD;
      v8f acc = {};
      for (int k = 0; k < DIMD; k += 4) {
        v2f a = *(const v2f*)(&sH[arow * XS + k + khalf]);
        v2f b = *(const v2f*)(w1 + k + khalf);
        acc = wmma_f32_k4(a, b, acc);
      }
      if (lane < 16) {
        float bias = bq_g[col];
#pragma unroll
        for (int i = 0; i < 8; ++i) sQ[i * XS + col] = acc[i] + bias;
      }
    }
    __syncthreads();

    // ---- phase 4+5: glimpse scores u = v_g . tanh(q + e_g), masked softmax -> sP ----
    {
      const int m = wave;
      const int bg = bbase + m;
      const float* eg = e_g + (size_t)bg * SRCL * DIMD;
      float u[8];
#pragma unroll
      for (int j = 0; j < 8; ++j) {
        const int l = lane + 32 * j;
        const float* er = eg + (size_t)l * DIMD;
        v4f acc = {};
        for (int d = 0; d < DIMD; d += 4) {
          v4f e4 = *(const v4f*)(er + d);
          v4f q4 = *(const v4f*)(&sQ[m * XS + d]);
          v4f vv = *(const v4f*)(&sVg[d]);
          acc.x += vv.x * tanh_fast(q4.x + e4.x);
          acc.y += vv.y * tanh_fast(q4.y + e4.y);
          acc.z += vv.z * tanh_fast(q4.z + e4.z);
          acc.w += vv.w * tanh_fast(q4.w + e4.w);
        }
        float uu = (acc.x + acc.y) + (acc.z + acc.w);
        bool masked = (sMask[m][l >> 5] >> (l & 31)) & 1u;
        u[j] = masked ? -__builtin_inff() : uu;
      }
      float mx = u[0];
#pragma unroll
      for (int j = 1; j < 8; ++j) mx = fmaxf(mx, u[j]);
      for (int off = 16; off; off >>= 1) mx = fmaxf(mx, __shfl_xor(mx, off, 32));
      float ex[8], sum = 0.f;
#pragma unroll
      for (int j = 0; j < 8; ++j) { ex[j] = exp2_fast(LOG2E * (u[j] - mx)); sum += ex[j]; }
      for (int off = 16; off; off >>= 1) sum += __shfl_xor(sum, off, 32);
      float rs = rcp_fast(sum);
#pragma unroll
      for (int j = 0; j < 8; ++j) sP[m * PS + lane + 32 * j] = ex[j] * rs;
    }
    __syncthreads();

    // ---- phase 6: g_l[m][d] = sum_l e_g[b][l][d] * p[m][l] ----
    {
      const int m = t >> 5;
      const int d0 = (t & 31) * 4;
      const float* eg = e_g + (size_t)(bbase + m) * SRCL * DIMD + d0;
      v4f acc = {};
      for (int l = 0; l < SRCL; ++l) {
        float p = sP[m * PS + l];
        v4f e4 = *(const v4f*)(eg + (size_t)l * DIMD);
        acc.x += e4.x * p; acc.y += e4.y * p; acc.z += e4.z * p; acc.w += e4.w * p;
      }
      *(v4f*)(&sG[m * XS + d0]) = acc;
    }
    __syncthreads();

    // ---- phase 7 (WMMA): q2 = g_l @ Wq_p^T + bq_p ----
    {
      const int col = wave * 16 + arow;
      const float* w1 = Wq_p + (size_t)col * DIMD;
      v8f acc = {};
      for (int k = 0; k < DIMD; k += 4) {
        v2f a = *(const v2f*)(&sG[arow * XS + k + khalf]);
        v2f b = *(const v2f*)(w1 + k + khalf);
        acc = wmma_f32_k4(a, b, acc);
      }
      if (lane < 16) {
        float bias = bq_p[col];
#pragma unroll
        for (int i = 0; i < 8; ++i) sQ[i * XS + col] = acc[i] + bias;
      }
    }
    __syncthreads();

    // ---- phase 8+9: pointer logits, log-softmax -> out, argmax -> sIdx ----
    {
      const int m = wave;
      const int bg = bbase + m;
      const float* ep = e_p + (size_t)bg * SRCL * DIMD;
      float lg[8];
#pragma unroll
      for (int j = 0; j < 8; ++j) {
        const int l = lane + 32 * j;
        const float* er = ep + (size_t)l * DIMD;
        v4f acc = {};
        for (int d = 0; d < DIMD; d += 4) {
          v4f e4 = *(const v4f*)(er + d);
          v4f q4 = *(const v4f*)(&sQ[m * XS + d]);
          v4f vv = *(const v4f*)(&sVp[d]);
          acc.x += vv.x * tanh_fast(q4.x + e4.x);
          acc.y += vv.y * tanh_fast(q4.y + e4.y);
          acc.z += vv.z * tanh_fast(q4.z + e4.z);
          acc.w += vv.w * tanh_fast(q4.w + e4.w);
        }
        float uu = (acc.x + acc.y) + (acc.z + acc.w);
        bool masked = (sMask[m][l >> 5] >> (l & 31)) & 1u;
        lg[j] = masked ? -__builtin_inff() : C_EXPL * tanh_fast(uu);
      }
      float mx = lg[0];
#pragma unroll
      for (int j = 1; j < 8; ++j) mx = fmaxf(mx, lg[j]);
      for (int off = 16; off; off >>= 1) mx = fmaxf(mx, __shfl_xor(mx, off, 32));
      float sum = 0.f;
#pragma unroll
      for (int j = 0; j < 8; ++j) sum += exp2_fast(LOG2E * (lg[j] - mx));
      for (int off = 16; off; off >>= 1) sum += __shfl_xor(sum, off, 32);
      float lse = mx + LN2 * log2_fast(sum);
      float* outrow = out_logp + ((size_t)bg * SRCL + step) * SRCL;
#pragma unroll
      for (int j = 0; j < 8; ++j) outrow[lane + 32 * j] = lg[j] - lse;
      // argmax, ties -> lowest index (matches jnp.argmax)
      float best = lg[0];
      int bidx = lane;
#pragma unroll
      for (int j = 1; j < 8; ++j) {
        int l = lane + 32 * j;
        if (lg[j] > best) { best = lg[j]; bidx = l; }
      }
      for (int off = 16; off; off >>= 1) {
        float ob = __shfl_xor(best, off, 32);
        int oi = __shfl_xor(bidx, off, 32);
        if (ob > best || (ob == best && oi < bidx)) { best = ob; bidx = oi; }
      }
      if (lane == 0) {
        sIdx[m] = bidx;
        out_sels[(size_t)bg * SRCL + step] = (float)bidx;
      }
    }
    __syncthreads();

    // ---- phase 10: gather next decoder input x = embedded_inputs[idx][b] ----
    {
      const int m = t >> 5;
      const int d0 = (t & 31) * 4;
      const int bg = bbase + m;
      v4f e4 = *(const v4f*)(embedded_inputs +
                             ((size_t)sIdx[m] * BATCH + bg) * DIMD + d0);
      *(v4f*)(&sX[m * XS + d0]) = e4;
    }
    __syncthreads();
  }

  // ---- final h, c ----
  for (int i = t; i < BT * DIMD; i += NTHREADS) {
    int m = i >> 7, d = i & (DIMD - 1);
    out_h[(size_t)(bbase + m) * DIMD + d] = sH[m * XS + d];
    out_c[(size_t)(bbase + m) * DIMD + d] = sC[m * XS + d];
  }
}

// ---------------- host launcher ----------------
extern "C" void kernel_launch(void* const* d_in, const int* in_sizes, int n_in,
                              void* d_out, int out_size, void* d_ws, size_t ws_size,
                              hipStream_t stream) {
  const float* decoder_input   = (const float*)d_in[0];
  const float* embedded_inputs = (const float*)d_in[1];
  const float* h0   = (const float*)d_in[2];
  const float* c0   = (const float*)d_in[3];
  const float* context = (const float*)d_in[4];
  const float* W_ih = (const float*)d_in[5];
  const float* W_hh = (const float*)d_in[6];
  const float* b_ih = (const float*)d_in[7];
  const float* b_hh = (const float*)d_in[8];
  const float* Wq_g = (const float*)d_in[9];
  const float* bq_g = (const float*)d_in[10];
  const float* Wr_g = (const float*)d_in[11];
  const float* br_g = (const float*)d_in[12];
  const float* v_g  = (const float*)d_in[13];
  const float* Wq_p = (const float*)d_in[14];
  const float* bq_p = (const float*)d_in[15];
  const float* Wr_p = (const float*)d_in[16];
  const float* br_p = (const float*)d_in[17];
  const float* v_p  = (const float*)d_in[18];

  // workspace: e_g (64MB) + e_p (64MB), both L2-resident on MI455X (192MB L2)
  float* e_g = (float*)d_ws;
  float* e_p = e_g + (size_t)BATCH * SRCL * DIMD;

  float* out_logp = (float*)d_out;
  float* out_sels = out_logp + (size_t)BATCH * SRCL * SRCL;
  float* out_h    = out_sels + (size_t)BATCH * SRCL;
  float* out_c    = out_h + (size_t)BATCH * DIMD;

  const int ewaves = 2 * (DIMD / 16) * ((BATCH * SRCL) / 16);   // 131072
  precompute_e_kernel<<<ewaves / (NTHREADS / 32), NTHREADS, 0, stream>>>(
      Wr_g, br_g, Wr_p, br_p, context, e_g, e_p);

  decoder_kernel<<<BATCH / BT, NTHREADS, 0, stream>>>(
      decoder_input, embedded_inputs, h0, c0,
      W_ih, W_hh, b_ih, b_hh,
      Wq_g, bq_g, v_g, Wq_p, bq_p, v_p,
      e_g, e_p, out_logp, out_sels, out_h, out_c);
}